// MahalanobisSimilarity_71502615544487
// MI455X (gfx1250) — compile-verified
//
#include <hip/hip_runtime.h>
#include <math.h>

#define B_ 4096
#define E_ 64
#define D_ 1024
#define KC 128   // D-chunk staged in LDS per iteration

typedef __attribute__((ext_vector_type(2))) float v2f;
typedef __attribute__((ext_vector_type(8))) float v8f;

// ---------------- prep: sigma_inv = exp(-log_sigma) ----------------
__global__ void sinv_kernel(const float* __restrict__ log_sigma,
                            float* __restrict__ sinv) {
    int d = blockIdx.x * 256 + threadIdx.x;
    if (d < D_) sinv[d] = __expf(-log_sigma[d]);
}

// ---------------- prep: kk[e] = sum_d sinv[d]*k[e,d]^2 ----------------
__global__ void kk_kernel(const float* __restrict__ keys,
                          const float* __restrict__ sinv,
                          float* __restrict__ kk) {
    __shared__ float red[256];
    const int e = blockIdx.x;
    float acc = 0.f;
    for (int d = threadIdx.x; d < D_; d += 256) {
        float k = keys[e * D_ + d];
        acc += sinv[d] * k * k;
    }
    red[threadIdx.x] = acc;
    __syncthreads();
    for (int s = 128; s > 0; s >>= 1) {
        if (threadIdx.x < s) red[threadIdx.x] += red[threadIdx.x + s];
        __syncthreads();
    }
    if (threadIdx.x == 0) kk[e] = red[0];
}

// ---------------- fused main kernel ----------------
// block = 128 threads (4 waves), handles 16 rows of z.
// GEMM1: cross[16,64] = (z*sinv)[16,D] @ keys^T[D,64]  (wave w -> experts 16w..16w+15)
// then sim, softmax, GEMM2: out[16,1024] = w[16,64] @ keys[64,1024]
__global__ __launch_bounds__(128) void
moe_kernel(const float* __restrict__ z, const float* __restrict__ keys,
           const float* __restrict__ sinv, const float* __restrict__ kk,
           float* __restrict__ out_sim, float* __restrict__ out_we) {
    __shared__ float zs[16][KC + 4];   // z*sinv chunk, padded stride 132
    __shared__ float kl[64][KC + 4];   // keys chunk
    __shared__ float zz[16];
    __shared__ float sw[16][68];       // sim, then softmax weights

    const int t    = threadIdx.x;
    const int wave = t >> 5;
    const int lane = t & 31;
    const int half = lane >> 4;        // A/B fragment K-split per ISA layout
    const int lid  = lane & 15;
    const int rowbase = blockIdx.x * 16;
    const int nb1 = wave * 16;         // expert N-tile for GEMM1

    if (t < 16) zz[t] = 0.f;
    __syncthreads();

    v8f c1 = {};
    for (int ch = 0; ch < D_ / KC; ++ch) {
        const int kb0 = ch * KC;
        // stage z*sinv tile; accumulate zz = sum sinv*z^2
        {
            const int r  = t >> 3;
            const int c0 = (t & 7) * 16;
            const float* zp = z + (size_t)(rowbase + r) * D_ + kb0 + c0;
            const float* sp = sinv + kb0 + c0;
            float part = 0.f;
#pragma unroll
            for (int j = 0; j < 16; ++j) {
                float zr = zp[j];
                float v  = zr * sp[j];
                zs[r][c0 + j] = v;
                part += zr * v;           // z * (sinv*z)
            }
            atomicAdd(&zz[r], part);
            if (ch + 1 < D_ / KC) __builtin_prefetch(zp + KC, 0, 0);
        }
        // stage keys chunk [64][128]
        {
            const int er = t >> 1;
            const int c0 = (t & 1) * 64;
            const float* kp = keys + (size_t)er * D_ + kb0 + c0;
#pragma unroll 8
            for (int j = 0; j < 64; ++j) kl[er][c0 + j] = kp[j];
        }
        __syncthreads();
        // GEMM1 K-loop over this chunk: 32 x V_WMMA_F32_16X16X4_F32
#pragma unroll 4
        for (int k4 = 0; k4 < KC / 4; ++k4) {
            const int col = k4 * 4 + 2 * half;
            v2f a = *(const v2f*)&zs[lid][col];          // A[M=lid][K pair]
            v2f b = *(const v2f*)&kl[nb1 + lid][col];    // B[K pair][N=lid] = keys[N][K]
            c1 = __builtin_amdgcn_wmma_f32_16x16x4_f32(
                     false, a, false, b, (short)0, c1, false, false);
        }
        __syncthreads();
    }

    // similarity = 1/(1 + zz - 2*cross + kk)
    const float kkv = kk[nb1 + lid];
#pragma unroll
    for (int v = 0; v < 8; ++v) {
        const int m = v + 8 * half;                      // C layout: VGPR v -> row m
        float mah = zz[m] - 2.f * c1[v] + kkv;
        float sim = 1.f / (1.f + mah);
        sw[m][nb1 + lid] = sim;
        out_sim[(size_t)(rowbase + m) * E_ + nb1 + lid] = sim;
    }
    __syncthreads();

    // row softmax over E=64 (16 rows, one thread each; trivial vs GEMM cost)
    if (t < 16) {
        float mx = -1e30f;
#pragma unroll
        for (int e = 0; e < E_; ++e) mx = fmaxf(mx, sw[t][e]);
        float s = 0.f;
#pragma unroll
        for (int e = 0; e < E_; ++e) s += __expf(sw[t][e] - mx);
        float inv = 1.f / s;
#pragma unroll
        for (int e = 0; e < E_; ++e) sw[t][e] = __expf(sw[t][e] - mx) * inv;
    }
    __syncthreads();

    // GEMM2: out[16,1024] = weights[16,64] @ keys[64,1024]; keys streamed from L2
    for (int nt = wave; nt < D_ / 16; nt += 4) {
        const int nb = nt * 16;
        v8f c2 = {};
#pragma unroll
        for (int ks = 0; ks < E_ / 4; ++ks) {
            const int kr = ks * 4 + 2 * half;            // expert index pair
            v2f a = *(const v2f*)&sw[lid][kr];           // A = weights[M=lid][K pair]
            v2f b;
            b.x = keys[(size_t)kr * D_ + nb + lid];      // B = keys[K][N=lid]
            b.y = keys[(size_t)(kr + 1) * D_ + nb + lid];
            c2 = __builtin_amdgcn_wmma_f32_16x16x4_f32(
                     false, a, false, b, (short)0, c2, false, false);
        }
#pragma unroll
        for (int v = 0; v < 8; ++v) {
            const int m = v + 8 * half;
            out_we[(size_t)(rowbase + m) * D_ + nb + lid] = c2[v];
        }
    }
}

extern "C" void kernel_launch(void* const* d_in, const int* in_sizes, int n_in,
                              void* d_out, int out_size, void* d_ws, size_t ws_size,
                              hipStream_t stream) {
    const float* z    = (const float*)d_in[0];
    const float* keys = (const float*)d_in[1];
    const float* ls   = (const float*)d_in[2];
    float* sinv = (float*)d_ws;            // D floats
    float* kkp  = sinv + D_;               // E floats
    float* out_sim = (float*)d_out;                       // [B,E]
    float* out_we  = out_sim + (size_t)B_ * E_;           // [B,D]

    sinv_kernel<<<(D_ + 255) / 256, 256, 0, stream>>>(ls, sinv);
    kk_kernel<<<E_, 256, 0, stream>>>(keys, sinv, kkp);
    moe_kernel<<<B_ / 16, 128, 0, stream>>>(z, keys, sinv, kkp, out_sim, out_we);
}